// sdfBoundaryModule_47021301957208
// MI455X (gfx1250) — compile-verified
//
#include <hip/hip_runtime.h>
#include <float.h>

// CDNA5 / gfx1250 : wave32, WMMA 16x16x4 f32.
typedef float v2f __attribute__((ext_vector_type(2)));
typedef float v8f __attribute__((ext_vector_type(8)));

#define NVERT 32

__device__ __forceinline__ float kernel_k_dev(float dr) {
    const float L2 = 0.69314718055994530942f;
    const float INV6PI = 1.0f / (6.0f * 3.14159265358979323846f);
    float ad = fabsf(dr);
    bool small = ad < 1e-5f;
    float d = small ? 1.0f : ad;
    float d2 = d * d;
    float d3 = d * d2;
    float d5 = d3 * d2;
    float d7 = d5 * d2;
    float srt = sqrtf(fmaxf(1.0f - d2, 1e-12f));
    float acd = acosf(fminf(fmaxf(d, -1.0f), 1.0f - 1e-7f));
    float integral = ((-15.0f * d7 - 126.0f * d5) * logf(2.0f * srt + 2.0f)
                      + (15.0f * d7 + 126.0f * d5) * logf(d)
                      + 6.0f * acd + 15.0f * L2 * d7
                      + srt * (97.0f * d5 + 60.0f * d3 - 22.0f * d)
                      + 126.0f * L2 * d5) * INV6PI;
    integral = (dr < 0.0f) ? (1.0f - integral) : integral;
    integral = small ? 0.5f : integral;
    float g = 1.0f - dr * 0.5f;
    float gamma = 1.0f + g * g;
    return integral * gamma;
}

// One wave = 16 points x 32 edges x 5 FD evaluations.
// Three rank-2 products (p.e, p.poly, cross(p,e)) via V_WMMA_F32_16X16X4_F32,
// computed once and shared across the 5 finite-difference evaluations through
// per-edge constant shifts. Edge tiles processed back-to-back to keep only
// 24 accumulator VGPRs live.
__global__ __launch_bounds__(256) void sdf_wmma_kernel(
        const float* __restrict__ poly,   // [32,2]
        const float* __restrict__ p,      // [M,2]
        float* __restrict__ out,          // [c:M | grad:2M | k:M | gk:2M]
        int M) {
    // Epilogue constants, 12 floats per edge (3x float4):
    // [0]=dh*ex [1]=dh*ey [2]=2dh*ax [3]=2dh*ay | [4]=ay [5]=polyNext.y [6]=ee [7]=cve | [8]=c2 [9]=c3 [10]=1/ee [11]=pad
    __shared__ float ec[NVERT * 12];
    // Raw edge data for A-matrix build: [ex, ey, ax, ay]
    __shared__ float ea[NVERT * 4];

    const int tid  = threadIdx.x;
    const int lane = tid & 31;
    const int wave = tid >> 5;
    const float DH = 1e-4f;

    if (tid < NVERT) {
        int n  = tid;
        int n1 = (tid + 1) & (NVERT - 1);
        float ax = poly[2 * n],  ay = poly[2 * n + 1];
        float bx = poly[2 * n1], by = poly[2 * n1 + 1];
        float ex = bx - ax, ey = by - ay;
        float ee = ex * ex + ey * ey;
        float* e = &ec[n * 12];
        e[0]  = DH * ex;                  // dhex
        e[1]  = DH * ey;                  // dhey
        e[2]  = 2.0f * DH * ax;           // t2px
        e[3]  = 2.0f * DH * ay;           // t2py
        e[4]  = ay;                       // cond1 threshold
        e[5]  = by;                       // poly[n+1].y (cond2)
        e[6]  = ee;
        e[7]  = -(ax * ex + ay * ey);     // cve : ve = pe + cve + o.e
        e[8]  = ax * ax + ay * ay;        // c2  : |poly|^2
        e[9]  = -(ex * ay - ey * ax);     // c3  : -cross(e, poly)
        e[10] = (ee > 0.0f) ? (1.0f / ee) : 0.0f;
        e[11] = 0.0f;
        float* a = &ea[n * 4];
        a[0] = ex; a[1] = ey; a[2] = ax; a[3] = ay;
    }
    __syncthreads();

    const int tile   = blockIdx.x * 8 + wave;
    const int nTiles = (M + 15) >> 4;
    if (tile >= nTiles) return;            // wave-uniform exit (EXEC all-1 for WMMA)

    const int  pt     = tile * 16 + (lane & 15);
    const int  ptc    = min(pt, M - 1);
    const bool loHalf = lane < 16;

    const float pxv = p[2 * ptc];
    const float pyv = p[2 * ptc + 1];

    // ---- B matrix (4x16, K x points). Interleaved-K layout: VGPR0 = K0|K2, VGPR1 = K1|K3.
    v2f Bm;
    Bm.x = loHalf ? pxv : 0.0f;            // K0 row = p.x ; K2 row = 0
    Bm.y = loHalf ? pyv : 0.0f;            // K1 row = p.y ; K3 row = 0

    // ---- per-lane point constants for the 5 evaluations
    const float dh2 = DH * DH;
    const float pp  = pxv * pxv + pyv * pyv;
    const float sArr[5]  = {pp,
                            pp + 2.0f * DH * pxv + dh2, pp - 2.0f * DH * pxv + dh2,
                            pp + 2.0f * DH * pyv + dh2, pp - 2.0f * DH * pyv + dh2};
    const float pyArr[5] = {pyv, pyv, pyv, pyv + DH, pyv - DH};

    float dmin[5] = {FLT_MAX, FLT_MAX, FLT_MAX, FLT_MAX, FLT_MAX};
    int   wn[5]   = {0, 0, 0, 0, 0};

#pragma unroll
    for (int t = 0; t < 2; ++t) {
        // A matrices (16x4, edges x K) for this edge tile.
        const float4 ar = *(const float4*)&ea[(t * 16 + (lane & 15)) * 4];
        const float aex = loHalf ? ar.x : 0.0f;
        const float aey = loHalf ? ar.y : 0.0f;
        const float aax = loHalf ? ar.z : 0.0f;
        const float aay = loHalf ? ar.w : 0.0f;
        const v2f Ape = (v2f){aex, aey};   // p.e
        const v2f Apv = (v2f){aax, aay};   // p.poly
        const v2f Apx = (v2f){aey, -aex};  // cross(p,e) = p.x*ey - p.y*ex

        const v8f z = {};
        v8f accPE = __builtin_amdgcn_wmma_f32_16x16x4_f32(false, Ape, false, Bm, (short)0, z, false, false);
        v8f accPV = __builtin_amdgcn_wmma_f32_16x16x4_f32(false, Apv, false, Bm, (short)0, z, false, false);
        v8f accPX = __builtin_amdgcn_wmma_f32_16x16x4_f32(false, Apx, false, Bm, (short)0, z, false, false);

        const int eBase = t * 16 + (loHalf ? 0 : 8);   // C layout: vgpr r -> M = r (lo) / 8+r (hi)
#pragma unroll
        for (int r = 0; r < 8; ++r) {
            const float4 k0 = *(const float4*)&ec[(eBase + r) * 12];
            const float4 k1 = *(const float4*)&ec[(eBase + r) * 12 + 4];
            const float4 k2 = *(const float4*)&ec[(eBase + r) * 12 + 8];
            const float dhex = k0.x, dhey = k0.y, t2px = k0.z, t2py = k0.w;
            const float ayv = k1.x, pny = k1.y, ee = k1.z, cve = k1.w;
            const float cc2 = k2.x, c3 = k2.y, iee = k2.z;

            const float pe  = accPE[r];
            const float pv  = accPV[r];
            const float pxr = accPX[r];

            const float peb   = pe + cve;
            const float term2 = fmaf(-2.0f, pv, cc2);   // -2 p.poly + |poly|^2

            const float oeA[5] = {0.0f,  dhex, -dhex,  dhey, -dhey};   // o.e
            const float c2o[5] = {0.0f, -t2px,  t2px, -t2py,  t2py};   // -2 o.poly
            const float ceo[5] = {0.0f, -dhey,  dhey,  dhex, -dhex};   // cross(e,o)
#pragma unroll
            for (int ev = 0; ev < 5; ++ev) {
                float ve = peb + oeA[ev];
                float tc = fminf(fmaxf(ve * iee, 0.0f), 1.0f);
                float vv = sArr[ev] + (term2 + c2o[ev]);               // |v|^2
                float d2 = fmaf(tc, fmaf(tc, ee, -2.0f * ve), vv);     // |v - t e|^2
                dmin[ev] = fminf(dmin[ev], d2);
                float val3 = (c3 + ceo[ev]) - pxr;                     // cross(e, v)
                bool c1b = pyArr[ev] >= ayv;
                bool c2b = pyArr[ev] < pny;
                wn[ev] += (c1b && c2b && (val3 > 0.0f)) ? 1
                        : ((!c1b) && (!c2b) && (val3 < 0.0f)) ? -1 : 0;
            }
            if (r & 1) __builtin_amdgcn_sched_barrier(0);  // cap load hoisting / live ranges
        }
    }

    // ---- pair reduce (lane L <-> L+16 hold the same point, complementary edges)
    float cEval[5];
#pragma unroll
    for (int ev = 0; ev < 5; ++ev) {
        float dm = fminf(dmin[ev], __shfl_xor(dmin[ev], 16, 32));
        int   w  = wn[ev] + __shfl_xor(wn[ev], 16, 32);
        float dd = sqrtf(fmaxf(dm, 0.0f));
        cEval[ev] = (w != 0) ? -dd : dd;
    }

    // ---- finite-difference gradient + kernel integral
    const float c0v = cEval[0];
    float gx = (cEval[1] - cEval[2]) * (0.5f / DH);
    float gy = (cEval[3] - cEval[4]) * (0.5f / DH);
    float gl = sqrtf(gx * gx + gy * gy);
    bool  gok = fabsf(gl) > 1e-5f;
    float ginv = gok ? (1.0f / gl) : 1.0f;
    gx = gok ? gx * ginv : gx;
    gy = gok ? gy * ginv : gy;

    const float SUP = 0.5f, KH = 0.01f;
    bool  mask = (c0v / SUP) <= 1.0f;
    float dr   = fminf(fmaxf(mask ? (c0v / SUP) : 0.5f, 0.0f), 1.0f);
    float xc   = kernel_k_dev(dr);
    float xr   = kernel_k_dev(fminf(fmaxf(dr + KH, 0.0f), 1.0f));
    float kg   = (xr - xc) / KH / SUP;
    float kv   = mask ? xc : 0.0f;
    float kgv  = mask ? kg : 0.0f;

    if (loHalf && pt < M) {
        out[pt] = c0v;
        out[M + 2 * pt]     = gx;
        out[M + 2 * pt + 1] = gy;
        out[3 * M + pt]     = kv;
        out[4 * M + 2 * pt]     = gx * kgv;
        out[4 * M + 2 * pt + 1] = gy * kgv;
    }
}

extern "C" void kernel_launch(void* const* d_in, const int* in_sizes, int n_in,
                              void* d_out, int out_size, void* d_ws, size_t ws_size,
                              hipStream_t stream) {
    const float* poly = (const float*)d_in[0];   // [32,2] f32
    const float* p    = (const float*)d_in[1];   // [M,2]  f32
    float* out = (float*)d_out;                  // 6*M f32
    const int M = in_sizes[1] / 2;
    const int nTiles = (M + 15) / 16;
    const int blocks = (nTiles + 7) / 8;         // 8 waves / 256-thread block
    sdf_wmma_kernel<<<blocks, 256, 0, stream>>>(poly, p, out, M);
}